// ParametricDrumSynth_17179869184062
// MI455X (gfx1250) — compile-verified
//
#include <hip/hip_runtime.h>
#include <stdint.h>

// ---------------------------------------------------------------------------
// ParametricDrumSynth for MI455X (gfx1250, wave32).
//
// Strategy: the three one-pole IIRs are made parallel via windowed exact
// re-scan (geometric forgetting: a^2048 <= 0.99^2048 ~ 1.2e-9 < f32 noise),
// and each 256-sample chunk scan is computed on the f32 matrix pipe as
// Y = L @ X with L the 16x16 lower-triangular Toeplitz filter matrix,
// using 4x V_WMMA_F32_16X16X4_F32 accumulation steps. Carries between the
// 16 segments / between chunks are a 16-step shuffle recurrence.
// ---------------------------------------------------------------------------

typedef __attribute__((ext_vector_type(2))) float v2f;
typedef __attribute__((ext_vector_type(8))) float v8f;

#define N_SAMPLES 8388608L
#define HALF_N    4194304u

__device__ __forceinline__ uint32_t rotl32(uint32_t x, int r) {
  return (x << r) | (x >> (32 - r));
}

// JAX threefry2x32 with key (0, keylo); counter pair (c0, c1); sel picks word.
__device__ __forceinline__ uint32_t threefry2x32(uint32_t keylo, uint32_t c0,
                                                 uint32_t c1, int sel) {
  const uint32_t ks0 = 0u, ks1 = keylo, ks2 = 0x1BD11BDAu ^ keylo;
  uint32_t x0 = c0 + ks0, x1 = c1 + ks1;
#define TF_R(r) { x0 += x1; x1 = rotl32(x1, r); x1 ^= x0; }
  TF_R(13) TF_R(15) TF_R(26) TF_R(6)
  x0 += ks1; x1 += ks2 + 1u;
  TF_R(17) TF_R(29) TF_R(16) TF_R(24)
  x0 += ks2; x1 += ks0 + 2u;
  TF_R(13) TF_R(15) TF_R(26) TF_R(6)
  x0 += ks0; x1 += ks1 + 3u;
  TF_R(17) TF_R(29) TF_R(16) TF_R(24)
  x0 += ks1; x1 += ks2 + 4u;
  TF_R(13) TF_R(15) TF_R(26) TF_R(6)
  x0 += ks2; x1 += ks0 + 5u;
#undef TF_R
  return sel ? x1 : x0;
}

// XLA's f32 ErfInv polynomial (what jax.random.normal lowers to).
__device__ __forceinline__ float erfinv32(float x) {
  float w = -log1pf(-x * x);
  float p;
  if (w < 5.0f) {
    w -= 2.5f;
    p = 2.81022636e-08f;
    p = fmaf(p, w, 3.43273939e-07f);
    p = fmaf(p, w, -3.5233877e-06f);
    p = fmaf(p, w, -4.39150654e-06f);
    p = fmaf(p, w, 0.00021858087f);
    p = fmaf(p, w, -0.00125372503f);
    p = fmaf(p, w, -0.00417768164f);
    p = fmaf(p, w, 0.246640727f);
    p = fmaf(p, w, 1.50140941f);
  } else {
    w = sqrtf(w) - 3.0f;
    p = -0.000200214257f;
    p = fmaf(p, w, 0.000100950558f);
    p = fmaf(p, w, 0.00134934322f);
    p = fmaf(p, w, -0.00367342844f);
    p = fmaf(p, w, 0.00573950773f);
    p = fmaf(p, w, -0.0076224613f);
    p = fmaf(p, w, 0.00943887047f);
    p = fmaf(p, w, 1.00167406f);
    p = fmaf(p, w, 2.83297682f);
  }
  return p * x;
}

// Reproduces jax.random.normal(jax.random.key(keylo), (2^23,), f32)[idx].
// idx < 0 returns 0 (pre-history / zero initial condition).
__device__ __forceinline__ float tf_normal(long idx, uint32_t keylo) {
  if (idx < 0) return 0.0f;
  uint32_t i; int sel;
  if (idx < (long)HALF_N) { i = (uint32_t)idx;                 sel = 0; }
  else                    { i = (uint32_t)(idx - (long)HALF_N); sel = 1; }
  uint32_t bits = threefry2x32(keylo, i, i + HALF_N, sel);
  bits = (bits >> 9) | 0x3f800000u;
  float f = __uint_as_float(bits) - 1.0f;           // [0,1)
  float u = fmaf(f, 1.99999994f, -0.99999994f);     // [-1+eps, 1)
  u = fmaxf(u, -0.99999994f);
  return 1.41421356f * erfinv32(u);
}

__device__ __forceinline__ float powi(float a, int e) {
  float r = 1.0f;
  for (int k = 0; k < e; ++k) r *= a;
  return r;
}

// Build the WMMA A-matrix registers for L[i][j] = (j<=i) ? c*a^(i-j) : 0.
// A layout (16x4 per K-chunk): lanes 0-15 hold M=lane, K={4kb,4kb+1};
// lanes 16-31 hold M=lane-16, K={4kb+2,4kb+3}.
__device__ __forceinline__ void make_filter(v2f A[4], float c, float a,
                                            float &a16, float &rowbase) {
  const int lane = threadIdx.x & 31;
  const int i  = lane & 15;
  const int jh = (lane >> 4) & 1;
#pragma unroll
  for (int kb = 0; kb < 4; ++kb) {
    const int j0 = 4 * kb + 2 * jh;
    A[kb][0] = (j0     <= i) ? c * powi(a, i - j0    ) : 0.0f;
    A[kb][1] = (j0 + 1 <= i) ? c * powi(a, i - j0 - 1) : 0.0f;
  }
  a16 = powi(a, 16);
  rowbase = a * (jh ? powi(a, 8) : 1.0f);   // a^{row0+1}, row0 = 0 or 8
}

// Exact scan of one 256-sample chunk. ld(off) supplies x[chunk_base + off].
// carry holds y[-1] on entry and y[255] on exit. Returns this lane's 8 output
// rows (column s = lane&15; rows 0..7 for lanes<16, rows 8..15 otherwise).
template <typename Ld>
__device__ __forceinline__ v8f scan_chunk(const v2f A[4], float a, float a16,
                                          float rowbase, float &carry, Ld &&ld) {
  const int lane = threadIdx.x & 31;
  const int s  = lane & 15;
  const int jh = (lane >> 4) & 1;

  v8f d = {0, 0, 0, 0, 0, 0, 0, 0};
#pragma unroll
  for (int kb = 0; kb < 4; ++kb) {
    const int j0 = 4 * kb + 2 * jh;
    v2f b;
    b[0] = ld(s * 16 + j0);
    b[1] = ld(s * 16 + j0 + 1);
    d = __builtin_amdgcn_wmma_f32_16x16x4_f32(false, A[kb], false, b,
                                              (short)0, d, false, false);
  }

  // Segment carries: P_0 = carry_in, P_{r+1} = a^16 * P_r + Ylocal[15][r].
  // Ylocal row 15 lives in d[7] of lanes 16..31 (column = lane-16).
  const float last = d[7];
  float P = carry, Pm = carry;
#pragma unroll
  for (int r = 0; r < 16; ++r) {
    const float lr = __shfl(last, 16 + r, 32);
    if (r == s) Pm = P;
    P = fmaf(a16, P, lr);
  }
  carry = P;  // y[255] (identical across all lanes)

  v8f y;
  float coef = rowbase;  // a^{i+1} for this lane's first row
#pragma unroll
  for (int r = 0; r < 8; ++r) {
    y[r] = fmaf(coef, Pm, d[r]);
    coef *= a;
  }
  return y;
}

__global__ __launch_bounds__(256)
void drum_synth_kernel(const float *__restrict__ prm, float *__restrict__ out) {
  __shared__ float slp2[4096];  // lp2 over [wgbase-2048, wgbase+2048)

  const int tid  = threadIdx.x;
  const int wave = tid >> 5;
  const int lane = tid & 31;
  const int s    = lane & 15;
  const int jh   = (lane >> 4) & 1;
  const long wgbase = (long)blockIdx.x * 2048;

  const float decay_t  = prm[1];
  const float freq     = prm[2];
  const float satur    = prm[3];
  const float gain_t   = prm[4];
  const float attack_n = prm[8];
  const float decay_n  = prm[9];
  const float c_lp     = prm[10];
  const float c_hp     = prm[11];
  const float gain_n   = prm[12];

  const float a1 = 0.9f, a2 = 1.0f - c_lp, a3 = 1.0f - c_hp;
  v2f A1[4], A2[4], A3[4];
  float a1_16, a2_16, a3_16, rb1, rb2, rb3;
  make_filter(A1, 0.1f, a1, a1_16, rb1);
  make_filter(A2, c_lp, a2, a2_16, rb2);
  make_filter(A3, c_hp, a3, a3_16, rb3);

  // ---- Phase 1: lp2 = lowpass(randn(key2), c_lp) -> LDS -------------------
  // Each wave produces 512 samples (4 warm-up chunks: a2^1024 <= 3.7e-24).
  {
    float carry = 0.0f;
    const long gout0 = wgbase - 2048 + (long)wave * 512;
    for (int ci = 0; ci < 6; ++ci) {
      const long cb = gout0 - 1024 + (long)ci * 256;
      v8f y = scan_chunk(A2, a2, a2_16, rb2, carry,
                         [&](int off) { return tf_normal(cb + off, 2u); });
      if (ci >= 4) {
        float *p = &slp2[wave * 512 + (ci - 4) * 256 + s * 16 + jh * 8];
        ((float4 *)p)[0] = make_float4(y[0], y[1], y[2], y[3]);
        ((float4 *)p)[1] = make_float4(y[4], y[5], y[6], y[7]);
      }
    }
  }
  __syncthreads();

  const long obase = wgbase + (long)wave * 256;

  // ---- Phase 2a: lp3 = lowpass(lp2, c_hp), 2048-sample warm-up from LDS ---
  v8f y3 = {0, 0, 0, 0, 0, 0, 0, 0};
  {
    float carry = 0.0f;
    const int lbase = wave * 256;
    for (int ci = 0; ci < 9; ++ci) {
      const int lcb = lbase + ci * 256;
      v8f y = scan_chunk(A3, a3, a3_16, rb3, carry,
                         [&](int off) { return slp2[lcb + off]; });
      if (ci == 8) y3 = y;
    }
  }

  // ---- Phase 2b: lp1 = lowpass(randn(key1), 0.1), 512 warm-up -------------
  v8f y1 = {0, 0, 0, 0, 0, 0, 0, 0};
  {
    float carry = 0.0f;
    for (int ci = 0; ci < 3; ++ci) {
      const long cb = obase - 512 + (long)ci * 256;
      v8f y = scan_chunk(A1, a1, a1_16, rb1, carry,
                         [&](int off) { return tf_normal(cb + off, 1u); });
      if (ci == 2) y1 = y;
    }
  }

  // ---- Phase 2c: transient + envelopes, fused store -----------------------
  const float inv_nm1 = 1.0f / 8388607.0f;
  const float omega = 6.2831853071795864f * freq;
  const float d20 = 20.0f * decay_t;
  const float ghalf = 0.5f * gain_t;
  const int  loff = wave * 256 + 2048 + s * 16 + jh * 8;
  const long g0   = obase + s * 16 + jh * 8;

  float r8[8];
#pragma unroll
  for (int r = 0; r < 8; ++r) {
    const float t = (float)(g0 + r) * inv_nm1;
    const float hp = slp2[loff + r] - y3[r];                 // highpass out
    float sine = sinf(omega * t) * expf(-decay_t * t) * gain_t;
    const float trans = tanhf(sine * satur);
    const float nt = y1[r] * expf(-d20 * t) * ghalf;
    const float env = (1.0f - expf(-attack_n * t)) * expf(-decay_n * t);
    r8[r] = trans + nt + hp * env * gain_n;
  }
  float4 *op = (float4 *)(out + g0);
  op[0] = make_float4(r8[0], r8[1], r8[2], r8[3]);
  op[1] = make_float4(r8[4], r8[5], r8[6], r8[7]);
}

extern "C" void kernel_launch(void *const *d_in, const int *in_sizes, int n_in,
                              void *d_out, int out_size, void *d_ws,
                              size_t ws_size, hipStream_t stream) {
  (void)in_sizes; (void)n_in; (void)out_size; (void)d_ws; (void)ws_size;
  const float *prm = (const float *)d_in[0];
  float *out = (float *)d_out;
  // 2^23 samples / 2048 per workgroup = 4096 workgroups of 8 waves.
  drum_synth_kernel<<<dim3(4096), dim3(256), 0, stream>>>(prm, out);
}